// SimpleEmbeddedAtomPotential_12128987644536
// MI455X (gfx1250) — compile-verified
//
#include <hip/hip_runtime.h>
#include <hip/hip_bf16.h>

// ---------------------------------------------------------------------------
// SimpleEmbeddedAtomPotential: fused forward energy + analytic VJP forces.
// Memory/atomic-bound scatter-gather workload (no matrix ops -> no WMMA).
// gfx1250 paths used: native global_atomic_add_f32, ds_add_f32, wave32,
// async global->LDS copy (ASYNCcnt) in the node pass.
// ---------------------------------------------------------------------------

#if defined(__has_builtin)
#if __has_builtin(__builtin_amdgcn_global_load_async_to_lds_b64) && \
    __has_builtin(__builtin_amdgcn_s_wait_asynccnt)
#define EAM_ASYNC_LDS 1
#endif
#endif

#if EAM_ASYNC_LDS
// Builtin signature (decoded from clang diagnostics):
//   void __builtin_amdgcn_global_load_async_to_lds_b64(
//       v2i __device__* gaddr, v2i __shared__* laddr, imm offset, imm cpol)
typedef int eam_v2i __attribute__((vector_size(2 * sizeof(int))));
typedef __attribute__((address_space(1))) eam_v2i* eam_gptr_v2i;
typedef __attribute__((address_space(3))) eam_v2i* eam_lptr_v2i;
#endif

__device__ __forceinline__ float softplus_f(float x) {
    // matches jax.nn.softplus = max(x,0) + log1p(exp(-|x|))
    return fmaxf(x, 0.0f) + log1pf(expf(-fabsf(x)));
}

__device__ __forceinline__ void gatomic_add(float* p, float v) {
    // Hardware global_atomic_add_f32 (no CAS loop) — confirmed by histogram.
    unsafeAtomicAdd(p, v);
}

// Pass 1: per-edge density & repulsion, scatter-sum onto dst nodes.
// dens_rep is interleaved [2*N]: {density, rep} per node -> both atomics hit
// the same cacheline.
__global__ void eam_edge_fwd(const float* __restrict__ r,
                             const int*   __restrict__ dst,
                             const float* __restrict__ p_da,
                             const float* __restrict__ p_dl,
                             const float* __restrict__ p_ra,
                             const float* __restrict__ p_rl,
                             float* __restrict__ dens_rep,
                             int E)
{
    const float A_d = softplus_f(*p_da);
    const float l_d = softplus_f(*p_dl);
    const float A_r = softplus_f(*p_ra);
    const float l_r = softplus_f(*p_rl);

    int i = blockIdx.x * blockDim.x + threadIdx.x;
    if (i >= E) return;

    const float x = r[3 * i + 0];
    const float y = r[3 * i + 1];
    const float z = r[3 * i + 2];
    const float b = sqrtf(fmaf(x, x, fmaf(y, y, z * z)));

    const float dens = A_d * expf(-l_d * b);
    const float rep  = A_r * expf(-l_r * b);

    const int d = dst[i];
    gatomic_add(&dens_rep[2 * d + 0], dens);
    gatomic_add(&dens_rep[2 * d + 1], rep);
}

// Pass 2: per-node embedding F = -A_e*sqrt(max(rho,1e-12)), gradient
// coefficient g_ld = dE/d(rho), and per-graph energy reduction
// (F + rep_node) via LDS ds_add_f32 then G global atomics per block.
// The {rho, rep} pair is staged global->LDS with the CDNA5 async copy path.
__global__ void eam_node_pass(const float* __restrict__ dens_rep,
                              const int*   __restrict__ node2graph,
                              const float* __restrict__ p_ea,
                              float* __restrict__ g_ld,
                              float* __restrict__ e_out,
                              int N, int G)
{
    __shared__ float sE[256];
#if EAM_ASYNC_LDS
    __shared__ float s_dr[2 * 256];
#endif

    const float A_e = softplus_f(*p_ea);
    const int tid = threadIdx.x;

    for (int j = tid; j < G; j += blockDim.x) sE[j] = 0.0f;
    __syncthreads();

    const int i = blockIdx.x * blockDim.x + tid;

#if EAM_ASYNC_LDS
    if (i < N) {
        // async copy this thread's {rho, rep} pair into its private LDS slot
        __builtin_amdgcn_global_load_async_to_lds_b64(
            (eam_gptr_v2i)(dens_rep + 2 * (size_t)i),
            (eam_lptr_v2i)(s_dr + 2 * tid),
            /*offset=*/0, /*cpol=*/0);
    }
    __builtin_amdgcn_s_wait_asynccnt(0);  // drain this wave's ASYNCcnt
#endif

    if (i < N) {
#if EAM_ASYNC_LDS
        const float rho = s_dr[2 * tid + 0];
        const float rpn = s_dr[2 * tid + 1];
#else
        const float rho = dens_rep[2 * i + 0];
        const float rpn = dens_rep[2 * i + 1];
#endif
        const float F = -A_e * sqrtf(fmaxf(rho, 1e-12f));
        // d/d(rho) of -A_e*sqrt(max(rho,eps)); clamp kills grad below eps.
        g_ld[i] = (rho > 1e-12f) ? (-0.5f * A_e / sqrtf(rho)) : 0.0f;
        const int g = node2graph[i];
        atomicAdd(&sE[g], F + rpn);      // LDS ds_add_f32
    }
    __syncthreads();

    for (int j = tid; j < G; j += blockDim.x) {
        gatomic_add(&e_out[j], sE[j]);
    }
}

// Pass 3: analytic backward. Recompute per-edge terms (r/dst stream hits the
// 192MB L2 from pass 1), form dE/dr, scatter +/- into forces.
__global__ void eam_edge_bwd(const float* __restrict__ r,
                             const int*   __restrict__ src,
                             const int*   __restrict__ dst,
                             const float* __restrict__ g_ld,
                             const float* __restrict__ p_da,
                             const float* __restrict__ p_dl,
                             const float* __restrict__ p_ra,
                             const float* __restrict__ p_rl,
                             float* __restrict__ forces,
                             int E)
{
    const float A_d = softplus_f(*p_da);
    const float l_d = softplus_f(*p_dl);
    const float A_r = softplus_f(*p_ra);
    const float l_r = softplus_f(*p_rl);

    int i = blockIdx.x * blockDim.x + threadIdx.x;
    if (i >= E) return;

    const float x = r[3 * i + 0];
    const float y = r[3 * i + 1];
    const float z = r[3 * i + 2];
    const float b = sqrtf(fmaf(x, x, fmaf(y, y, z * z)));

    const float dens = A_d * expf(-l_d * b);
    const float rep  = A_r * expf(-l_r * b);

    const int d = dst[i];
    const int s = src[i];

    // dE/d(bondlen) = g_rho[dst] * d(dens)/db + 1 * d(rep)/db
    const float gb = g_ld[d] * (-l_d * dens) - l_r * rep;
    const float sc = gb / b;                 // d(bondlen)/dr = r / b
    const float fx = sc * x;
    const float fy = sc * y;
    const float fz = sc * z;

    // forces = segsum(-dE_dr, dst) + segsum(+dE_dr, src)
    gatomic_add(&forces[3 * d + 0], -fx);
    gatomic_add(&forces[3 * d + 1], -fy);
    gatomic_add(&forces[3 * d + 2], -fz);
    gatomic_add(&forces[3 * s + 0],  fx);
    gatomic_add(&forces[3 * s + 1],  fy);
    gatomic_add(&forces[3 * s + 2],  fz);
}

extern "C" void kernel_launch(void* const* d_in, const int* in_sizes, int n_in,
                              void* d_out, int out_size, void* d_ws, size_t ws_size,
                              hipStream_t stream)
{
    const float* r    = (const float*)d_in[0];
    const int*   src  = (const int*)  d_in[1];
    const int*   dst  = (const int*)  d_in[2];
    const int*   n2g  = (const int*)  d_in[3];
    const float* p_da = (const float*)d_in[4];
    const float* p_dl = (const float*)d_in[5];
    const float* p_ra = (const float*)d_in[6];
    const float* p_rl = (const float*)d_in[7];
    const float* p_ea = (const float*)d_in[8];
    // d_in[9] = num_nodes, d_in[10] = num_graphs live on device; derive on host:
    const int E = in_sizes[1];
    const int N = in_sizes[3];
    const int G = out_size - 3 * N;   // output = [energy(G), forces(3N)]

    float* dens_rep = (float*)d_ws;                 // [2N] interleaved
    float* g_ld     = dens_rep + 2 * (size_t)N;     // [N]
    float* e_out    = (float*)d_out;                // [G]
    float* forces   = e_out + G;                    // [3N]

    // Atomic accumulators must start from zero every call.
    (void)hipMemsetAsync(d_ws, 0, 3 * (size_t)N * sizeof(float), stream);
    (void)hipMemsetAsync(d_out, 0, (size_t)out_size * sizeof(float), stream);

    const int B = 256;                 // 8 wave32's per workgroup
    const int gridE = (E + B - 1) / B;
    const int gridN = (N + B - 1) / B;

    eam_edge_fwd<<<gridE, B, 0, stream>>>(r, dst, p_da, p_dl, p_ra, p_rl,
                                          dens_rep, E);
    eam_node_pass<<<gridN, B, 0, stream>>>(dens_rep, n2g, p_ea, g_ld,
                                           e_out, N, G);
    eam_edge_bwd<<<gridE, B, 0, stream>>>(r, src, dst, g_ld,
                                          p_da, p_dl, p_ra, p_rl, forces, E);
}